// GCNNet_82205674045498
// MI455X (gfx1250) — compile-verified
//
#include <hip/hip_runtime.h>
#include <hip/hip_bf16.h>
#include <math.h>

typedef __attribute__((ext_vector_type(2))) float v2f;
typedef __attribute__((ext_vector_type(8))) float v8f;

#define F_INC   256
#define HDIM    256
#define DDIM    128
#define TEMPC   0.6f
#define EPSG    1e-20f

// ---------------------------------------------------------------------------
// degree / normalization
// ---------------------------------------------------------------------------
__global__ void init_deg_kernel(float* __restrict__ deg, int n) {
    int i = blockIdx.x * blockDim.x + threadIdx.x;
    if (i < n) deg[i] = 1.0f;          // +1 self loop
}

__global__ void deg_edges_kernel(const int* __restrict__ dst, float* __restrict__ deg, int e) {
    int i = blockIdx.x * blockDim.x + threadIdx.x;
    if (i < e) atomicAdd(&deg[dst[i]], 1.0f);
}

__global__ void dinv_kernel(const float* __restrict__ deg, float* __restrict__ dinv, int n) {
    int i = blockIdx.x * blockDim.x + threadIdx.x;
    if (i < n) dinv[i] = rsqrtf(deg[i]);
}

// ---------------------------------------------------------------------------
// f32 WMMA GEMM with fused GCN self-loop epilogue.
//   H[M,NC]   = A[M,K] @ B[K,NC]                  (h_pre / logits_pre)
//   AGG[M,NC] = dinv[m]^2 * H[m,n] + bias[n]      (aggregation init)
// Row-major. K, NC compile-time; K%4==0, NC%64==0.
//
// Block = 256 threads = 8 waves tiled 4(m) x 2(n); each wave owns a 16x32
// C strip (two 16x16 fragments sharing one A fragment) -> block tile 64x64.
// V_WMMA_F32_16X16X4_F32 lane layout:
//   A 16x4 : lanes 0-15 -> M=lane, K={k,k+1}; lanes 16-31 -> K={k+2,k+3}
//   B  4x16: mirrored (N = lane&15)
//   C      : vgpr r -> M = r + 8*(lane>>4), N = lane&15
// Out-of-range M rows: A row index clamped for loads (row m of A only feeds
// row m of D), stores guarded -> no predication inside the K loop.
// ---------------------------------------------------------------------------
template <int K, int NC>
__global__ __launch_bounds__(256) void wmma_gemm_gcn(
    const float* __restrict__ A, const float* __restrict__ B,
    float* __restrict__ H, float* __restrict__ AGG,
    const float* __restrict__ dinv, const float* __restrict__ bias, int M) {

    const int lane = threadIdx.x & 31;
    const int wave = threadIdx.x >> 5;
    const int wm   = wave & 3;
    const int wn   = wave >> 2;

    const int m0   = blockIdx.x * 64 + wm * 16;
    const int n0   = blockIdx.y * 64 + wn * 32;

    const int khalf = (lane >> 4) * 2;              // 0 | 2
    const int row   = m0 + (lane & 15);
    const int rowc  = (row < M) ? row : (M - 1);    // clamp: loads always legal
    const int col0  = n0 + (lane & 15);

    const float* ap  = A + (size_t)rowc * K + khalf;     // 8B-aligned
    const float* bp0 = B + (size_t)khalf * NC + col0;
    const float* bp1 = bp0 + 16;

    v8f c0 = {};
    v8f c1 = {};

#pragma unroll 8
    for (int k = 0; k < K; k += 4) {
        v2f a = *(const v2f*)(ap + k);               // global_load_b64
        v2f b0, b1;
        b0.x = bp0[(size_t)k * NC];
        b0.y = bp0[(size_t)k * NC + NC];
        b1.x = bp1[(size_t)k * NC];
        b1.y = bp1[(size_t)k * NC + NC];
        c0 = __builtin_amdgcn_wmma_f32_16x16x4_f32(false, a, false, b0,
                                                   (short)0, c0, false, false);
        c1 = __builtin_amdgcn_wmma_f32_16x16x4_f32(false, a, false, b1,
                                                   (short)0, c1, false, false);
    }

    const int mbase = m0 + (lane >> 4) * 8;
#pragma unroll
    for (int r = 0; r < 8; ++r) {
        const int rr = mbase + r;
        if (rr < M) {
            const float di  = dinv[rr];
            const float dsq = di * di;
            const size_t o0 = (size_t)rr * NC + col0;
            H[o0]        = c0[r];
            H[o0 + 16]   = c1[r];
            AGG[o0]      = dsq * c0[r] + bias[col0];
            AGG[o0 + 16] = dsq * c1[r] + bias[col0 + 16];
        }
    }
}

// ---------------------------------------------------------------------------
// Edge aggregation: agg[dst] += dinv[src]*dinv[dst] * h[src]   (float4/thread)
// F power of two; lgGroups = log2(F/4)
// ---------------------------------------------------------------------------
__global__ void edge_agg_kernel(const int* __restrict__ src, const int* __restrict__ dst,
                                const float* __restrict__ dinv, const float* __restrict__ h,
                                float* __restrict__ agg, int e, int F, int lgGroups) {
    long long t = (long long)blockIdx.x * blockDim.x + threadIdx.x;
    long long total = (long long)e << lgGroups;
    if (t >= total) return;
    int ed = (int)(t >> lgGroups);
    int fg = ((int)t & ((1 << lgGroups) - 1)) << 2;
    int s = src[ed];
    int d = dst[ed];
    float coef = dinv[s] * dinv[d];
    const float4 hv = *(const float4*)(h + (size_t)s * F + fg);
    float* ap = agg + (size_t)d * F + fg;
    atomicAdd(ap + 0, coef * hv.x);
    atomicAdd(ap + 1, coef * hv.y);
    atomicAdd(ap + 2, coef * hv.z);
    atomicAdd(ap + 3, coef * hv.w);
}

// ---------------------------------------------------------------------------
// Gumbel-softmax + straight-through hard one-hot. One wave32 per node,
// 4 features per lane (D=128). Matches jnp.argmax first-index tie-break.
// ---------------------------------------------------------------------------
__global__ __launch_bounds__(256) void gumbel_out_kernel(
    const float* __restrict__ logits, const float* __restrict__ U,
    float* __restrict__ out, int n) {

    const int lane = threadIdx.x & 31;
    const int wv   = threadIdx.x >> 5;
    const int node = blockIdx.x * 8 + wv;
    if (node >= n) return;

    const size_t base = (size_t)node * DDIM;
    float z[4];
#pragma unroll
    for (int j = 0; j < 4; ++j) {
        const int d = lane + 32 * j;
        float u = U[base + d];
        float g = -logf(-logf(u + EPSG) + EPSG);
        z[j] = (logits[base + d] + g) / TEMPC;
    }

    float m  = z[0];
    int   mi = lane;
#pragma unroll
    for (int j = 1; j < 4; ++j) {
        const int d = lane + 32 * j;
        if (z[j] > m) { m = z[j]; mi = d; }
    }
    for (int off = 16; off >= 1; off >>= 1) {
        float om  = __shfl_xor(m, off, 32);
        int   omi = __shfl_xor(mi, off, 32);
        if (om > m || (om == m && omi < mi)) { m = om; mi = omi; }
    }

    float e[4];
    float s = 0.0f;
#pragma unroll
    for (int j = 0; j < 4; ++j) { e[j] = expf(z[j] - m); s += e[j]; }
    for (int off = 16; off >= 1; off >>= 1) s += __shfl_xor(s, off, 32);
    const float inv = 1.0f / s;

#pragma unroll
    for (int j = 0; j < 4; ++j) {
        const int d = lane + 32 * j;
        float y = e[j] * inv;
        float hard = (d == mi) ? 1.0f : 0.0f;
        out[base + d] = (hard - y) + y;
    }
}

// ---------------------------------------------------------------------------
// launch
// ---------------------------------------------------------------------------
extern "C" void kernel_launch(void* const* d_in, const int* in_sizes, int n_in,
                              void* d_out, int out_size, void* d_ws, size_t ws_size,
                              hipStream_t stream) {
    const float* x   = (const float*)d_in[0];
    const int*   ei  = (const int*)d_in[1];
    const float* W1  = (const float*)d_in[2];
    const float* b1  = (const float*)d_in[3];
    const float* W2  = (const float*)d_in[4];
    const float* b2  = (const float*)d_in[5];
    const float* U   = (const float*)d_in[6];
    float* out = (float*)d_out;

    const int n = in_sizes[0] / F_INC;     // 100000
    const int e = in_sizes[1] / 2;         // 1600000
    const int* src = ei;
    const int* dst = ei + e;

    // workspace layout (floats)
    float* w = (float*)d_ws;
    float* deg  = w;                    w += n;
    float* dinv = w;                    w += n;
    float* bufA = w;                    w += (size_t)n * HDIM;   // h_pre, then logits_pre
    float* bufB = w;                    w += (size_t)n * HDIM;   // agg1 == h (SLOPE=1 -> identity)
    float* bufC = w;                    /* n*DDIM */             // agg2 == final logits

    const int T = 256;
    auto blocks = [](long long total, int t) { return (unsigned)((total + t - 1) / t); };

    // degrees + normalization
    init_deg_kernel<<<blocks(n, T), T, 0, stream>>>(deg, n);
    deg_edges_kernel<<<blocks(e, T), T, 0, stream>>>(dst, deg, e);
    dinv_kernel<<<blocks(n, T), T, 0, stream>>>(deg, dinv, n);

    // layer 1: h_pre = x @ W1 ; agg1 init fused (bufA = h_pre, bufB = dinv^2*h_pre + b1)
    {
        dim3 grid((n + 63) / 64, HDIM / 64);
        wmma_gemm_gcn<F_INC, HDIM><<<grid, 256, 0, stream>>>(x, W1, bufA, bufB, dinv, b1, n);
    }
    edge_agg_kernel<<<blocks((long long)e * (HDIM / 4), T), T, 0, stream>>>(
        src, dst, dinv, bufA, bufB, e, HDIM, 6 /* log2(256/4) */);
    // leaky_relu with negative_slope == 1.0 is the identity -> h = bufB

    // layer 2: logits_pre = h @ W2 ; agg2 init fused
    {
        dim3 grid((n + 63) / 64, DDIM / 64);
        wmma_gemm_gcn<HDIM, DDIM><<<grid, 256, 0, stream>>>(bufB, W2, bufA, bufC, dinv, b2, n);
    }
    edge_agg_kernel<<<blocks((long long)e * (DDIM / 4), T), T, 0, stream>>>(
        src, dst, dinv, bufA, bufC, e, DDIM, 5 /* log2(128/4) */);

    // gumbel-softmax straight-through output
    gumbel_out_kernel<<<blocks((long long)n * 32, T), T, 0, stream>>>(bufC, U, out, n);
}